// Transformer_52819507806815
// MI455X (gfx1250) — compile-verified
//
#include <hip/hip_runtime.h>
#include <hip/hip_bf16.h>
#include <math.h>

#define NNODES 50000
#define NEDGES 800000
#define DINC   128
#define HC     128   // H*C
#define EDIM   32

typedef __attribute__((ext_vector_type(2))) float v2f;
typedef __attribute__((ext_vector_type(4))) float v4f;
typedef __attribute__((ext_vector_type(8))) float v8f;

__device__ __forceinline__ void atomicMaxF32(float* addr, float val) {
    // float ordering == int ordering for non-negative, reversed-unsigned for negative
    if (val >= 0.0f) atomicMax((int*)addr, __float_as_int(val));
    else             atomicMin((unsigned int*)addr, __float_as_uint(val));
}

// ---------------------------------------------------------------------------
// K0: init workspace accumulators (required every call: harness poisons d_ws)
// ---------------------------------------------------------------------------
__global__ void gt_init_kernel(float* __restrict__ accum,
                               float* __restrict__ mmax,
                               float* __restrict__ denom) {
    int i = blockIdx.x * blockDim.x + threadIdx.x;
    if (i < NNODES * HC) accum[i] = 0.0f;
    if (i < NNODES * 2) {
        mmax[i]  = __int_as_float(0xFF800000); // -inf
        denom[i] = 0.0f;
    }
}

// ---------------------------------------------------------------------------
// K1: fused node projections  q|k|v|skip = x @ W + b   via V_WMMA_F32_16X16X4
// One wave computes one 16x16 output tile; K-loop of 32 wmma ops (K=128).
// A frag (16x4 f32): lanes 0-15 hold K=k0,k0+1 of row M=lane; lanes 16-31 K=k0+2,k0+3
// B frag (4x16 f32): lanes 0-15 hold rows K=k0,k0+1 at col N=lane; lanes 16-31 rows K=k0+2,k0+3
// C (16x16 f32): VGPR i: lanes 0-15 -> M=i, N=lane; lanes 16-31 -> M=i+8, N=lane-16
// ---------------------------------------------------------------------------
__global__ void gt_node_proj_kernel(const float* __restrict__ x,
                                    const float* __restrict__ Wq, const float* __restrict__ bq,
                                    const float* __restrict__ Wk, const float* __restrict__ bk,
                                    const float* __restrict__ Wv, const float* __restrict__ bv,
                                    const float* __restrict__ Ws, const float* __restrict__ bs,
                                    float* __restrict__ q, float* __restrict__ k,
                                    float* __restrict__ v, float* __restrict__ skp) {
    const int wave = (blockIdx.x * blockDim.x + threadIdx.x) >> 5;
    const int lane = threadIdx.x & 31;
    // wave -> (rowTile [0,3125), mat [0,4), colTile [0,8))
    const int rowTile = wave >> 5;
    const int rem     = wave & 31;
    const int mat     = rem >> 3;
    const int colTile = rem & 7;

    const float* W; const float* b; float* out;
    switch (mat) {
        case 0:  W = Wq; b = bq; out = q;   break;
        case 1:  W = Wk; b = bk; out = k;   break;
        case 2:  W = Wv; b = bv; out = v;   break;
        default: W = Ws; b = bs; out = skp; break;
    }

    const int rowBase = rowTile * 16;
    const int colBase = colTile * 16;
    const int nn      = lane & 15;   // N within tile (and M-row for A loads)
    const int khalf   = lane >> 4;   // 0: K lo pair, 1: K hi pair

    const float* xrow = x + (size_t)(rowBase + nn) * DINC + 2 * khalf;

    v8f acc = {};
    #pragma unroll
    for (int k0 = 0; k0 < DINC; k0 += 4) {
        v2f a = *(const v2f*)(xrow + k0);                 // A[M=nn][k0+2*khalf .. +1]
        const int kk = k0 + 2 * khalf;
        v2f bb;
        bb.x = W[(size_t)kk       * HC + colBase + nn];    // B[kk][N=nn]
        bb.y = W[(size_t)(kk + 1) * HC + colBase + nn];    // B[kk+1][N=nn]
        acc = __builtin_amdgcn_wmma_f32_16x16x4_f32(
                  /*neg_a=*/false, a, /*neg_b=*/false, bb,
                  /*c_mod=*/(short)0, acc, /*reuse_a=*/false, /*reuse_b=*/false);
    }

    const float bias = b[colBase + nn];
    #pragma unroll
    for (int i = 0; i < 8; ++i) {
        const int row = rowBase + i + khalf * 8;
        out[(size_t)row * HC + colBase + nn] = acc[i] + bias;
    }
}

// ---------------------------------------------------------------------------
// K2: per-edge attention logit + segment max.
// One wave per edge; lane owns 4 channels (lanes 0-15 head0, 16-31 head1).
// e-feature recomputed on the fly from edge_attr @ We (We cache-resident).
// ---------------------------------------------------------------------------
__global__ void gt_edge_alpha_kernel(const int* __restrict__ ei,
                                     const float* __restrict__ ea,
                                     const float* __restrict__ We,
                                     const float* __restrict__ q,
                                     const float* __restrict__ k,
                                     float* __restrict__ alpha,
                                     float* __restrict__ mmax) {
    const int e    = (blockIdx.x * blockDim.x + threadIdx.x) >> 5;
    const int lane = threadIdx.x & 31;
    const int src = ei[e];
    const int dst = ei[NEDGES + e];
    const int c0  = lane * 4;

    const float eal = ea[(size_t)e * EDIM + lane];  // coalesced; one value per lane
    float ef0 = 0.f, ef1 = 0.f, ef2 = 0.f, ef3 = 0.f;
    #pragma unroll
    for (int j = 0; j < EDIM; ++j) {
        const float aj = __shfl(eal, j, 32);
        const v4f w = *(const v4f*)(We + (size_t)j * HC + c0);
        ef0 += aj * w.x; ef1 += aj * w.y; ef2 += aj * w.z; ef3 += aj * w.w;
    }

    const v4f qv = *(const v4f*)(q + (size_t)dst * HC + c0);
    const v4f kv = *(const v4f*)(k + (size_t)src * HC + c0);
    float s = qv.x * (kv.x + ef0) + qv.y * (kv.y + ef1)
            + qv.z * (kv.z + ef2) + qv.w * (kv.w + ef3);

    // reduce within each 16-lane half (one half per head)
    #pragma unroll
    for (int off = 1; off < 16; off <<= 1) s += __shfl_xor(s, off, 32);
    s *= 0.125f;   // 1/sqrt(C), C=64

    if ((lane & 15) == 0) {
        const int h = lane >> 4;
        alpha[(size_t)e * 2 + h] = s;
        atomicMaxF32(&mmax[(size_t)dst * 2 + h], s);
    }
}

// ---------------------------------------------------------------------------
// K3: per-edge exp, denominator accumulation, and unnormalized aggregation.
// ---------------------------------------------------------------------------
__global__ void gt_edge_agg_kernel(const int* __restrict__ ei,
                                   const float* __restrict__ ea,
                                   const float* __restrict__ We,
                                   const float* __restrict__ v,
                                   const float* __restrict__ alpha,
                                   const float* __restrict__ mmax,
                                   float* __restrict__ denom,
                                   float* __restrict__ accum) {
    const int e    = (blockIdx.x * blockDim.x + threadIdx.x) >> 5;
    const int lane = threadIdx.x & 31;
    const int src = ei[e];
    const int dst = ei[NEDGES + e];
    const int h   = lane >> 4;
    const int c0  = lane * 4;

    const float a = alpha[(size_t)e * 2 + h];
    const float p = __expf(a - mmax[(size_t)dst * 2 + h]);
    if ((lane & 15) == 0) atomicAdd(&denom[(size_t)dst * 2 + h], p);

    const float eal = ea[(size_t)e * EDIM + lane];
    float ef0 = 0.f, ef1 = 0.f, ef2 = 0.f, ef3 = 0.f;
    #pragma unroll
    for (int j = 0; j < EDIM; ++j) {
        const float aj = __shfl(eal, j, 32);
        const v4f w = *(const v4f*)(We + (size_t)j * HC + c0);
        ef0 += aj * w.x; ef1 += aj * w.y; ef2 += aj * w.z; ef3 += aj * w.w;
    }

    const v4f vv = *(const v4f*)(v + (size_t)src * HC + c0);
    float* o = accum + (size_t)dst * HC + c0;
    atomicAdd(o + 0, (vv.x + ef0) * p);
    atomicAdd(o + 1, (vv.y + ef1) * p);
    atomicAdd(o + 2, (vv.z + ef2) * p);
    atomicAdd(o + 3, (vv.w + ef3) * p);
}

// ---------------------------------------------------------------------------
// K4: out = skip + accum / (denom + eps)   (full overwrite of d_out)
// ---------------------------------------------------------------------------
__global__ void gt_finalize_kernel(const float* __restrict__ skp,
                                   const float* __restrict__ accum,
                                   const float* __restrict__ denom,
                                   float* __restrict__ out) {
    const int i = blockIdx.x * blockDim.x + threadIdx.x;
    if (i >= NNODES * HC) return;
    const int n = i >> 7;
    const int h = (i & 127) >> 6;
    out[i] = skp[i] + accum[i] / (denom[(size_t)n * 2 + h] + 1e-16f);
}

extern "C" void kernel_launch(void* const* d_in, const int* in_sizes, int n_in,
                              void* d_out, int out_size, void* d_ws, size_t ws_size,
                              hipStream_t stream) {
    const float* x  = (const float*)d_in[0];
    const int*   ei = (const int*)  d_in[1];
    const float* ea = (const float*)d_in[2];
    const float* Wq = (const float*)d_in[3];
    const float* bq = (const float*)d_in[4];
    const float* Wk = (const float*)d_in[5];
    const float* bk = (const float*)d_in[6];
    const float* Wv = (const float*)d_in[7];
    const float* bv = (const float*)d_in[8];
    const float* We = (const float*)d_in[9];
    const float* Ws = (const float*)d_in[10];
    const float* bs = (const float*)d_in[11];
    float* out = (float*)d_out;

    // workspace layout (floats):
    float* ws    = (float*)d_ws;
    float* q     = ws;                                 // N*HC
    float* k     = q     + (size_t)NNODES * HC;        // N*HC
    float* v     = k     + (size_t)NNODES * HC;        // N*HC
    float* skp   = v     + (size_t)NNODES * HC;        // N*HC
    float* accum = skp   + (size_t)NNODES * HC;        // N*HC
    float* alpha = accum + (size_t)NNODES * HC;        // E*2
    float* mmax  = alpha + (size_t)NEDGES * 2;         // N*2
    float* denom = mmax  + (size_t)NNODES * 2;         // N*2
    // total ~135 MB

    const int B = 256;
    gt_init_kernel<<<(NNODES * HC + B - 1) / B, B, 0, stream>>>(accum, mmax, denom);
    // 3125 row tiles * 4 matrices * 8 col tiles = 100000 waves = 12500 blocks of 8 waves
    gt_node_proj_kernel<<<12500, B, 0, stream>>>(x, Wq, bq, Wk, bk, Wv, bv, Ws, bs,
                                                 q, k, v, skp);
    // one wave per edge: 800000 waves = 100000 blocks
    gt_edge_alpha_kernel<<<NEDGES / 8, B, 0, stream>>>(ei, ea, We, q, k, alpha, mmax);
    gt_edge_agg_kernel<<<NEDGES / 8, B, 0, stream>>>(ei, ea, We, v, alpha, mmax, denom, accum);
    gt_finalize_kernel<<<(NNODES * HC + B - 1) / B, B, 0, stream>>>(skp, accum, denom, out);
}